// WarpCanonical_11991548690891
// MI455X (gfx1250) — compile-verified
//
#include <hip/hip_runtime.h>
#include <hip/hip_bf16.h>

#define N_RAYS    2048
#define N_SAMPLES 32
#define NPTS      (N_RAYS * N_SAMPLES)   // 65536
#define NVERTS    6890
#define VPAD      6912                   // 432 tiles of 16 (divisible by 4 for unrolling)
#define NTILES    (VPAD / 16)            // 432

typedef __attribute__((ext_vector_type(2))) float v2f;
typedef __attribute__((ext_vector_type(8))) float v8f;

// workspace layout (in floats)
#define WS_TINV 0                        // 6890 * 16 floats (row-major inverse transforms)
#define WS_VBS  (NVERTS * 16)            // VPAD * 4 floats, per-vertex {k0, k2, k1, k3}
#define WS_IDX  (WS_VBS + VPAD * 4)      // NPTS ints (argmin vertex per point)

// ---------------------------------------------------------------------------
// Kernel 1: per-vertex prep: 4x4 inverse (adjugate) + WMMA-B-operand table.
// B column for vertex v is (k0,k1,k2,k3) = (-2vx,-2vy,-2vz,|v|^2) so that
// D[m][n] = A[m]·B[:,n] = -2 p·v + |v|^2 (|p|^2 dropped: constant per row).
// Stored as {k0,k2,k1,k3} so lane<16 loads (k0,k2) and lane>=16 loads (k1,k3)
// each as one aligned float2 (matches 32-bit B VGPR layout: v0=K{0,1}, v1=K{2,3}).
// ---------------------------------------------------------------------------
__global__ void prep_kernel(const float* __restrict__ verts,
                            const float* __restrict__ Ts,
                            float* __restrict__ ws) {
    int v = blockIdx.x * blockDim.x + threadIdx.x;
    if (v >= VPAD) return;
    float* vbs = ws + WS_VBS;
    if (v < NVERTS) {
        float vx = verts[v * 3 + 0], vy = verts[v * 3 + 1], vz = verts[v * 3 + 2];
        vbs[v * 4 + 0] = -2.0f * vx;              // k0
        vbs[v * 4 + 1] = -2.0f * vz;              // k2
        vbs[v * 4 + 2] = -2.0f * vy;              // k1
        vbs[v * 4 + 3] = vx * vx + vy * vy + vz * vz; // k3

        float m[16];
#pragma unroll
        for (int i = 0; i < 16; ++i) m[i] = Ts[v * 16 + i];
        float inv[16];
        inv[0]  =  m[5]*m[10]*m[15] - m[5]*m[11]*m[14] - m[9]*m[6]*m[15] + m[9]*m[7]*m[14] + m[13]*m[6]*m[11] - m[13]*m[7]*m[10];
        inv[4]  = -m[4]*m[10]*m[15] + m[4]*m[11]*m[14] + m[8]*m[6]*m[15] - m[8]*m[7]*m[14] - m[12]*m[6]*m[11] + m[12]*m[7]*m[10];
        inv[8]  =  m[4]*m[9]*m[15]  - m[4]*m[11]*m[13] - m[8]*m[5]*m[15] + m[8]*m[7]*m[13] + m[12]*m[5]*m[11] - m[12]*m[7]*m[9];
        inv[12] = -m[4]*m[9]*m[14]  + m[4]*m[10]*m[13] + m[8]*m[5]*m[14] - m[8]*m[6]*m[13] - m[12]*m[5]*m[10] + m[12]*m[6]*m[9];
        inv[1]  = -m[1]*m[10]*m[15] + m[1]*m[11]*m[14] + m[9]*m[2]*m[15] - m[9]*m[3]*m[14] - m[13]*m[2]*m[11] + m[13]*m[3]*m[10];
        inv[5]  =  m[0]*m[10]*m[15] - m[0]*m[11]*m[14] - m[8]*m[2]*m[15] + m[8]*m[3]*m[14] + m[12]*m[2]*m[11] - m[12]*m[3]*m[10];
        inv[9]  = -m[0]*m[9]*m[15]  + m[0]*m[11]*m[13] + m[8]*m[1]*m[15] - m[8]*m[3]*m[13] - m[12]*m[1]*m[11] + m[12]*m[3]*m[9];
        inv[13] =  m[0]*m[9]*m[14]  - m[0]*m[10]*m[13] - m[8]*m[1]*m[14] + m[8]*m[2]*m[13] + m[12]*m[1]*m[10] - m[12]*m[2]*m[9];
        inv[2]  =  m[1]*m[6]*m[15]  - m[1]*m[7]*m[14]  - m[5]*m[2]*m[15] + m[5]*m[3]*m[14] + m[13]*m[2]*m[7]  - m[13]*m[3]*m[6];
        inv[6]  = -m[0]*m[6]*m[15]  + m[0]*m[7]*m[14]  + m[4]*m[2]*m[15] - m[4]*m[3]*m[14] - m[12]*m[2]*m[7]  + m[12]*m[3]*m[6];
        inv[10] =  m[0]*m[5]*m[15]  - m[0]*m[7]*m[13]  - m[4]*m[1]*m[15] + m[4]*m[3]*m[13] + m[12]*m[1]*m[7]  - m[12]*m[3]*m[5];
        inv[14] = -m[0]*m[5]*m[14]  + m[0]*m[6]*m[13]  + m[4]*m[1]*m[14] - m[4]*m[2]*m[13] - m[12]*m[1]*m[6]  + m[12]*m[2]*m[5];
        inv[3]  = -m[1]*m[6]*m[11]  + m[1]*m[7]*m[10]  + m[5]*m[2]*m[11] - m[5]*m[3]*m[10] - m[9]*m[2]*m[7]   + m[9]*m[3]*m[6];
        inv[7]  =  m[0]*m[6]*m[11]  - m[0]*m[7]*m[10]  - m[4]*m[2]*m[11] + m[4]*m[3]*m[10] + m[8]*m[2]*m[7]   - m[8]*m[3]*m[6];
        inv[11] = -m[0]*m[5]*m[11]  + m[0]*m[7]*m[9]   + m[4]*m[1]*m[11] - m[4]*m[3]*m[9]  - m[8]*m[1]*m[7]   + m[8]*m[3]*m[5];
        inv[15] =  m[0]*m[5]*m[10]  - m[0]*m[6]*m[9]   - m[4]*m[1]*m[10] + m[4]*m[2]*m[9]  + m[8]*m[1]*m[6]   - m[8]*m[2]*m[5];
        float det  = m[0]*inv[0] + m[1]*inv[4] + m[2]*inv[8] + m[3]*inv[12];
        float rdet = 1.0f / det;
#pragma unroll
        for (int i = 0; i < 16; ++i) ws[WS_TINV + v * 16 + i] = inv[i] * rdet;
    } else {
        // padding vertices: huge score -> never selected by argmin
        vbs[v * 4 + 0] = 0.0f;
        vbs[v * 4 + 1] = 0.0f;
        vbs[v * 4 + 2] = 0.0f;
        vbs[v * 4 + 3] = 3.0e38f;
    }
}

// ---------------------------------------------------------------------------
// Kernel 2: nearest-vertex argmin via V_WMMA_F32_16X16X4_F32.
// One wave = 16 points (rows). A layout (16x4 f32): lanes 0-15 hold K=0,1
// (px,py), lanes 16-31 hold K=2,3 (pz,1). Loop over 432 vertex tiles of 16,
// each WMMA yields a 16x16 score tile in 8 accumulator VGPRs, tracked with
// running min + index. Final 16-lane shfl_xor min-reduction per half-wave.
// ---------------------------------------------------------------------------
__global__ void argmin_kernel(const float* __restrict__ pts,
                              const float* __restrict__ ws,
                              int* __restrict__ idxout) {
    const int lane = threadIdx.x & 31;
    const int wave = (blockIdx.x * blockDim.x + threadIdx.x) >> 5;
    const int base = wave * 16;          // first point row of this wave
    const int col  = lane & 15;          // N (vertex) column within tile
    const bool hi  = lane >= 16;
    const int  m   = base + col;         // point row this lane contributes for A

    v2f a;
    if (!hi) { a.x = pts[m * 3 + 0]; a.y = pts[m * 3 + 1]; }  // K=0,1
    else     { a.x = pts[m * 3 + 2]; a.y = 1.0f;           }  // K=2,3

    const float2* vbs2 = reinterpret_cast<const float2*>(ws + WS_VBS);
    const int pairSel = hi ? 1 : 0;

    float minv[8];
    int   mini[8];
#pragma unroll
    for (int g = 0; g < 8; ++g) { minv[g] = 3.4e38f; mini[g] = 0; }

#pragma unroll 4
    for (int t = 0; t < NTILES; ++t) {
        const int vcol = t * 16 + col;
        float2 bb = vbs2[vcol * 2 + pairSel];   // one aligned b64 load per lane
        v2f b; b.x = bb.x; b.y = bb.y;
        v8f c = {};
        c = __builtin_amdgcn_wmma_f32_16x16x4_f32(
                /*neg_a=*/false, a, /*neg_b=*/false, b,
                /*c_mod=*/(short)0, c, /*reuse_a=*/false, /*reuse_b=*/false);
#pragma unroll
        for (int g = 0; g < 8; ++g) {
            bool lt = c[g] < minv[g];
            minv[g] = lt ? c[g] : minv[g];
            mini[g] = lt ? vcol : mini[g];
        }
    }

    // min+argmin reduction across the 16 lanes of each half-wave
    // (lanes 0-15 own rows 0-7 in vgpr g; lanes 16-31 own rows 8-15)
#pragma unroll
    for (int off = 8; off >= 1; off >>= 1) {
#pragma unroll
        for (int g = 0; g < 8; ++g) {
            float ov = __shfl_xor(minv[g], off, 32);
            int   oi = __shfl_xor(mini[g], off, 32);
            bool lt = ov < minv[g];
            minv[g] = lt ? ov : minv[g];
            mini[g] = lt ? oi : mini[g];
        }
    }
    if (lane == 0) {
#pragma unroll
        for (int g = 0; g < 8; ++g) idxout[base + g] = mini[g];
    } else if (lane == 16) {
#pragma unroll
        for (int g = 0; g < 8; ++g) idxout[base + 8 + g] = mini[g];
    }
}

// ---------------------------------------------------------------------------
// Kernel 3: apply gathered T^-1, finite-difference dirs along the 32-sample
// axis (exactly one wave32 per ray), normalize, flip, write both outputs.
// ---------------------------------------------------------------------------
__global__ void transform_kernel(const float* __restrict__ pts,
                                 const float* __restrict__ tinv,
                                 const int* __restrict__ idx,
                                 float* __restrict__ out) {
    const int n    = blockIdx.x * blockDim.x + threadIdx.x; // point id
    const int lane = threadIdx.x & 31;                      // == sample index (S=32)

    float px = pts[n * 3 + 0], py = pts[n * 3 + 1], pz = pts[n * 3 + 2];
    int v = idx[n];
    const float4* M4 = reinterpret_cast<const float4*>(tinv + v * 16);
    float4 r0 = M4[0], r1 = M4[1], r2 = M4[2];

    float hx = -px, hy = -py, hz = pz;                       // flipped homogeneous
    float cx = r0.x * hx + r0.y * hy + r0.z * hz + r0.w;
    float cy = r1.x * hx + r1.y * hy + r1.z * hz + r1.w;
    float cz = r2.x * hx + r2.y * hy + r2.z * hz + r2.w;

    // dirs[s] = can[s+1] - can[s];  dirs[31] = dirs[30]
    int nl = (lane < 31) ? lane + 1 : 31;
    float dx = __shfl(cx, nl, 32) - cx;
    float dy = __shfl(cy, nl, 32) - cy;
    float dz = __shfl(cz, nl, 32) - cz;
    int sl = (lane == 31) ? 30 : lane;
    dx = __shfl(dx, sl, 32);
    dy = __shfl(dy, sl, 32);
    dz = __shfl(dz, sl, 32);

    float nrm  = sqrtf(dx * dx + dy * dy + dz * dz);
    float rinv = 1.0f / fmaxf(nrm, 1e-12f);
    float ux = dx * rinv, uy = dy * rinv, uz = dz * rinv;

    // flip = (-1, -1, 1) applied to both outputs
    out[n * 3 + 0] = -cx;
    out[n * 3 + 1] = -cy;
    out[n * 3 + 2] =  cz;
    float* o2 = out + NPTS * 3;
    o2[n * 3 + 0] = -ux;
    o2[n * 3 + 1] = -uy;
    o2[n * 3 + 2] =  uz;
}

extern "C" void kernel_launch(void* const* d_in, const int* in_sizes, int n_in,
                              void* d_out, int out_size, void* d_ws, size_t ws_size,
                              hipStream_t stream) {
    const float* pts   = (const float*)d_in[0];  // rays_points_world (1,R,1,S,3)
    // d_in[1] = rays_directions_world (unused by the reference computation)
    const float* verts = (const float*)d_in[2];  // vertices_posed (1,V,3)
    const float* Ts    = (const float*)d_in[3];  // Ts (1,V,4,4)
    float* ws  = (float*)d_ws;
    float* out = (float*)d_out;
    int*   idx = (int*)(ws + WS_IDX);

    prep_kernel<<<(VPAD + 255) / 256, 256, 0, stream>>>(verts, Ts, ws);
    // NPTS/16 = 4096 waves, 8 waves (256 threads) per block -> 512 blocks
    argmin_kernel<<<(NPTS / 16) / 8, 256, 0, stream>>>(pts, ws, idx);
    transform_kernel<<<NPTS / 256, 256, 0, stream>>>(pts, ws + WS_TINV, idx, out);
}